// Decoder_27384711479321
// MI455X (gfx1250) — compile-verified
//
#include <hip/hip_runtime.h>
#include <hip/hip_bf16.h>
#include <math.h>

// ---------------------------------------------------------------------------
// MI455X (gfx1250) GRU decoder step.
// Bandwidth-bound (~515 MB f32 weights, ~4.2 GFLOP, ~22us @ 23.3 TB/s).
// Batch GEMVs (M=16) -> V_WMMA_F32_16X16X4_F32 tiles. Each wave computes NT
// adjacent 16x16 tiles so one activation (A) load feeds NT weight (B) loads,
// maximizing the share of vmem issue devoted to the HBM weight stream.
// ---------------------------------------------------------------------------

#define VOCAB  32000
#define EMBED  128
#define HIDDEN 2048
#define LAYERS 3
#define BATCH  16
#define G3H    (3 * HIDDEN)   // 6144

typedef float v2f __attribute__((ext_vector_type(2)));
typedef float v8f __attribute__((ext_vector_type(8)));

// ---------------------------------------------------------------------------
// Kernel 1: embedding lookup + ReLU.
// ---------------------------------------------------------------------------
__global__ void embed_relu_kernel(const long long* __restrict__ tokens,
                                  const float* __restrict__ emb,
                                  float* __restrict__ x) {
    int idx = blockIdx.x * blockDim.x + threadIdx.x;   // 0 .. 16*128-1
    if (idx < BATCH * EMBED) {
        int m = idx / EMBED;
        int e = idx - m * EMBED;
        long long t = tokens[m];
        float v = emb[t * (long long)EMBED + e];
        x[idx] = v > 0.0f ? v : 0.0f;
    }
}

// ---------------------------------------------------------------------------
// WMMA tile body: D[0:16, n0 : n0+NT*16] = X(16,K) @ W(N,K)^T + bias
// 256 threads = 8 waves; wave w owns K-slice [w*K/8,(w+1)*K/8) and NT
// accumulators; LDS cross-wave reduction at the end.
//
// 32-bit A (16x4) lane layout: lane l -> (m = l%16, k = k0 + 2*(l/16)),
// v0 = even k, v1 = odd k.  B (4x16) mirrors with n = ntile_base + l%16.
// ---------------------------------------------------------------------------
template <int K, int NT>
__device__ __forceinline__
void gemm16_tile_body(const float* __restrict__ X,
                      const float* __restrict__ W,
                      const float* __restrict__ bias,
                      float* __restrict__ D,
                      int n0, int ldD) {
    __shared__ float red[8][NT][BATCH * 16];

    constexpr int KSLICE = K / 8;                        // per-wave K
    constexpr int STEP   = (KSLICE % 32 == 0) ? 32 : 16; // k per unrolled iter
    constexpr int NSUB   = STEP / 4;                     // wmma steps per iter

    const int tid  = threadIdx.x;
    const int wave = tid >> 5;          // 0..7
    const int lane = tid & 31;          // 0..31

    const int mrow = lane & 15;         // m for A, tile-local n for B
    const int koff = 2 * (lane >> 4);   // 0 or 2

    const int kbeg = wave * KSLICE;
    const int kend = kbeg + KSLICE;

    const float* ap = X + mrow * K + koff;
    const float* bp[NT];
    #pragma unroll
    for (int t = 0; t < NT; ++t)
        bp[t] = W + (long long)(n0 + 16 * t + mrow) * K + koff;

    v8f acc[NT];
    #pragma unroll
    for (int t = 0; t < NT; ++t) acc[t] = (v8f){};

    for (int k0 = kbeg; k0 < kend; k0 += STEP) {
        v2f a[NSUB], b[NT][NSUB];
        #pragma unroll
        for (int u = 0; u < NSUB; ++u)
            a[u] = *(const v2f*)(ap + k0 + 4 * u);
        #pragma unroll
        for (int t = 0; t < NT; ++t)
            #pragma unroll
            for (int u = 0; u < NSUB; ++u)   // weights stream once: NT hint
                b[t][u] = __builtin_nontemporal_load((const v2f*)(bp[t] + k0 + 4 * u));
        #pragma unroll
        for (int u = 0; u < NSUB; ++u)
            #pragma unroll
            for (int t = 0; t < NT; ++t)     // interleave independent accs
                acc[t] = __builtin_amdgcn_wmma_f32_16x16x4_f32(
                    false, a[u], false, b[t][u], (short)0, acc[t], false, false);
    }

    // C/D layout: vgpr v -> lanes0-15: (M=v, N=lane), lanes16-31: (M=v+8).
    const int nloc  = lane & 15;
    const int mbase = (lane >> 4) * 8;
    #pragma unroll
    for (int t = 0; t < NT; ++t)
        #pragma unroll
        for (int v = 0; v < 8; ++v)
            red[wave][t][(mbase + v) * 16 + nloc] = acc[t][v];
    __syncthreads();

    // NT*256 outputs, 256 threads.
    #pragma unroll
    for (int t = tid; t < NT * BATCH * 16; t += 256) {
        const int tile = t >> 8;
        const int idx  = t & 255;
        float s = 0.0f;
        #pragma unroll
        for (int w = 0; w < 8; ++w) s += red[w][tile][idx];
        const int m = idx >> 4;
        const int n = n0 + tile * 16 + (idx & 15);
        s += bias[n];
        D[(long long)m * ldD + n] = s;
    }
}

template <int K, int NT>
__global__ void __launch_bounds__(256)
gemm16_kernel(const float* __restrict__ X, const float* __restrict__ W,
              const float* __restrict__ bias, float* __restrict__ D, int ldD) {
    gemm16_tile_body<K, NT>(X, W, bias, D, blockIdx.x * (NT * 16), ldD);
}

// Fused: blocks [0,ntiles0) compute set0 (gi), the rest compute set1 (gh).
template <int NT>
__global__ void __launch_bounds__(256)
gemm16_dual2048_kernel(const float* __restrict__ X0, const float* __restrict__ W0,
                       const float* __restrict__ b0, float* __restrict__ D0,
                       const float* __restrict__ X1, const float* __restrict__ W1,
                       const float* __restrict__ b1, float* __restrict__ D1,
                       int ntiles0, int ldD) {
    if ((int)blockIdx.x < ntiles0) {
        gemm16_tile_body<HIDDEN, NT>(X0, W0, b0, D0, blockIdx.x * (NT * 16), ldD);
    } else {
        gemm16_tile_body<HIDDEN, NT>(X1, W1, b1, D1,
                                     (blockIdx.x - ntiles0) * (NT * 16), ldD);
    }
}

// ---------------------------------------------------------------------------
// GRU gate elementwise math.
// ---------------------------------------------------------------------------
__global__ void gru_elem_kernel(const float* __restrict__ gi,   // (16, 6144)
                                const float* __restrict__ gh,   // (16, 6144)
                                const float* __restrict__ hprev,// (16, 2048)
                                float* __restrict__ hout) {     // (16, 2048)
    int idx = blockIdx.x * blockDim.x + threadIdx.x;   // 0 .. 16*2048-1
    if (idx >= BATCH * HIDDEN) return;
    int m = idx / HIDDEN;
    int j = idx - m * HIDDEN;
    const float* gim = gi + (long long)m * G3H;
    const float* ghm = gh + (long long)m * G3H;

    float ir = gim[j],              hr = ghm[j];
    float iz = gim[HIDDEN + j],     hz = ghm[HIDDEN + j];
    float in = gim[2 * HIDDEN + j], hn = ghm[2 * HIDDEN + j];

    float r = 1.0f / (1.0f + __expf(-(ir + hr)));
    float z = 1.0f / (1.0f + __expf(-(iz + hz)));
    float n = tanhf(in + r * hn);
    float h = hprev[idx];
    hout[idx] = (1.0f - z) * n + z * h;
}

// ---------------------------------------------------------------------------
// Row-wise log_softmax over 32000 logits (one block per batch row).
// ---------------------------------------------------------------------------
__global__ void __launch_bounds__(256)
log_softmax_kernel(const float* __restrict__ logits,  // (16, 32000)
                   float* __restrict__ out) {         // (16, 32000)
    __shared__ float s[256];
    const int m   = blockIdx.x;
    const int tid = threadIdx.x;
    const float* row = logits + (long long)m * VOCAB;
    float* orow      = out    + (long long)m * VOCAB;

    // pass 1: max
    float mx = -INFINITY;
    for (int j = tid; j < VOCAB; j += 256) mx = fmaxf(mx, row[j]);
    s[tid] = mx; __syncthreads();
    for (int off = 128; off > 0; off >>= 1) {
        if (tid < off) s[tid] = fmaxf(s[tid], s[tid + off]);
        __syncthreads();
    }
    mx = s[0]; __syncthreads();

    // pass 2: sum of exp
    float sum = 0.0f;
    for (int j = tid; j < VOCAB; j += 256) sum += __expf(row[j] - mx);
    s[tid] = sum; __syncthreads();
    for (int off = 128; off > 0; off >>= 1) {
        if (tid < off) s[tid] += s[tid + off];
        __syncthreads();
    }
    float lse = mx + __logf(s[0]);

    // pass 3: write
    for (int j = tid; j < VOCAB; j += 256) orow[j] = row[j] - lse;
}

// ---------------------------------------------------------------------------
// Host-side orchestration.
// d_in order: tokens, hidden, emb, {W_ih,W_hh,b_ih,b_hh} x3, W_out, b_out
// d_out: logp (16*32000) then new_hidden (3*16*2048)
// ---------------------------------------------------------------------------
extern "C" void kernel_launch(void* const* d_in, const int* in_sizes, int n_in,
                              void* d_out, int out_size, void* d_ws, size_t ws_size,
                              hipStream_t stream) {
    (void)in_sizes; (void)n_in; (void)out_size; (void)ws_size;

    const long long* tokens = (const long long*)d_in[0];
    const float* hidden = (const float*)d_in[1];
    const float* emb    = (const float*)d_in[2];
    const float* W_ih[LAYERS] = {(const float*)d_in[3], (const float*)d_in[7],  (const float*)d_in[11]};
    const float* W_hh[LAYERS] = {(const float*)d_in[4], (const float*)d_in[8],  (const float*)d_in[12]};
    const float* b_ih[LAYERS] = {(const float*)d_in[5], (const float*)d_in[9],  (const float*)d_in[13]};
    const float* b_hh[LAYERS] = {(const float*)d_in[6], (const float*)d_in[10], (const float*)d_in[14]};
    const float* W_out = (const float*)d_in[15];
    const float* b_out = (const float*)d_in[16];

    float* out   = (float*)d_out;
    float* logp  = out;                            // (16, 32000)
    float* new_h = out + (long long)BATCH * VOCAB; // (3, 16, 2048)

    // workspace layout (floats)
    float* ws     = (float*)d_ws;
    float* x0     = ws;                              // 16*128
    float* gi     = x0 + BATCH * EMBED;              // 16*6144
    float* gh     = gi + BATCH * G3H;                // 16*6144
    float* logits = gh + BATCH * G3H;                // 16*32000

    // 1) embedding + ReLU
    embed_relu_kernel<<<8, 256, 0, stream>>>(tokens, emb, x0);

    // 2) GRU layer 0 (K_ih = 128)
    {
        const float* h0 = hidden;
        float* h_out = new_h;
        gemm16_kernel<EMBED,  2><<<G3H / 32, 256, 0, stream>>>(x0, W_ih[0], b_ih[0], gi, G3H);
        gemm16_kernel<HIDDEN, 2><<<G3H / 32, 256, 0, stream>>>(h0, W_hh[0], b_hh[0], gh, G3H);
        gru_elem_kernel<<<(BATCH * HIDDEN + 255) / 256, 256, 0, stream>>>(gi, gh, h0, h_out);
    }

    // 3) GRU layers 1,2 (both GEMMs K = 2048 -> fused launch, NT=2)
    const float* layer_in = new_h;
    for (int l = 1; l < LAYERS; ++l) {
        const float* h_l = hidden + (long long)l * BATCH * HIDDEN;
        float* h_out = new_h + (long long)l * BATCH * HIDDEN;

        gemm16_dual2048_kernel<2><<<2 * (G3H / 32), 256, 0, stream>>>(
            layer_in, W_ih[l], b_ih[l], gi,
            h_l,      W_hh[l], b_hh[l], gh,
            G3H / 32, G3H);
        gru_elem_kernel<<<(BATCH * HIDDEN + 255) / 256, 256, 0, stream>>>(gi, gh, h_l, h_out);

        layer_in = h_out;
    }

    // 4) output projection (NT=4, 500 blocks) + log_softmax
    gemm16_kernel<HIDDEN, 4><<<VOCAB / 64, 256, 0, stream>>>(layer_in, W_out, b_out, logits, VOCAB);
    log_softmax_kernel<<<BATCH, 256, 0, stream>>>(logits, logp);
}